// GINModel_4947802325325
// MI455X (gfx1250) — compile-verified
//
#include <hip/hip_runtime.h>

typedef __attribute__((ext_vector_type(2))) float v2f;
typedef __attribute__((ext_vector_type(8))) float v8f;

#define N_NODES 100000
#define N_EDGES 1600000
#define IN_CH   128
#define HID     128
#define OUT_CH  64

// ---------------------------------------------------------------------------
// Zero the aggregation buffer (float4 stores; n divisible by 4*256).
// ---------------------------------------------------------------------------
__global__ __launch_bounds__(256) void zero_kernel(float* __restrict__ p, size_t n) {
    size_t i = ((size_t)blockIdx.x * blockDim.x + threadIdx.x) * 4;
    if (i + 3 < n) {
        float4 z = make_float4(0.f, 0.f, 0.f, 0.f);
        *(float4*)(p + i) = z;
    }
}

// ---------------------------------------------------------------------------
// Edge scatter-add: agg[dst] += h[src].  One wave per edge; lane l handles
// features [4l, 4l+4).  Gather is a coalesced 16B load (feature table is
// L2-resident: 51.2 MB << 192 MB); scatter is device-scope f32 atomic adds
// which lower to global_atomic_add_f32 resolved at L2.
// ---------------------------------------------------------------------------
__global__ __launch_bounds__(256) void scatter_kernel(
    const float* __restrict__ h,
    const int*   __restrict__ src,
    const int*   __restrict__ dst,
    float*       __restrict__ agg)
{
    int wave = blockIdx.x * (blockDim.x >> 5) + (threadIdx.x >> 5);
    int lane = threadIdx.x & 31;
    if (wave >= N_EDGES) return;

    int s = src[wave];
    int d = dst[wave];

    const float4 v = *(const float4*)(h + (size_t)s * HID + lane * 4);
    float* base = agg + (size_t)d * HID + lane * 4;

    __hip_atomic_fetch_add(base + 0, v.x, __ATOMIC_RELAXED, __HIP_MEMORY_SCOPE_AGENT);
    __hip_atomic_fetch_add(base + 1, v.y, __ATOMIC_RELAXED, __HIP_MEMORY_SCOPE_AGENT);
    __hip_atomic_fetch_add(base + 2, v.z, __ATOMIC_RELAXED, __HIP_MEMORY_SCOPE_AGENT);
    __hip_atomic_fetch_add(base + 3, v.w, __ATOMIC_RELAXED, __HIP_MEMORY_SCOPE_AGENT);
}

// ---------------------------------------------------------------------------
// OUT = act((X + AGG) @ W + bias) via V_WMMA_F32_16X16X4_F32.
// One wave -> one 16x16 tile of OUT.  K stepped by 4.
//
// A (16x4 f32, 2 VGPRs): lanes 0-15 carry rows M=0..15 with K=k+0,k+1;
//                        lanes 16-31 carry the same rows with K=k+2,k+3.
// B (4x16 f32, 2 VGPRs): mirrored — lane%16 = column N, lane/2^4 selects
//                        K-pair {0,1} vs {2,3}.
// C/D (16x16 f32, 8 VGPRs): VGPR r, lanes 0-15 -> (M=r, N=lane),
//                           lanes 16-31 -> (M=r+8, N=lane-16).
// Bias is per-column => broadcast bias[n] into all 8 accumulator VGPRs.
// ---------------------------------------------------------------------------
template <int KDIM, int NDIM, bool RELU>
__global__ __launch_bounds__(256) void gin_gemm_kernel(
    const float* __restrict__ X,     // [N_NODES, KDIM]
    const float* __restrict__ AGG,   // [N_NODES, KDIM]
    const float* __restrict__ W,     // [KDIM, NDIM]
    const float* __restrict__ bias,  // [NDIM]
    float*       __restrict__ OUT)   // [N_NODES, NDIM]
{
    constexpr int COL_TILES = NDIM / 16;

    int waveId  = blockIdx.x * (blockDim.x >> 5) + (threadIdx.x >> 5);
    int rowTile = waveId / COL_TILES;
    int colTile = waveId % COL_TILES;
    if (rowTile >= N_NODES / 16) return;

    const int m0   = rowTile * 16;
    const int n0   = colTile * 16;
    const int lane = threadIdx.x & 31;
    const int half = lane >> 4;      // 0: K-pair {0,1}; 1: K-pair {2,3}
    const int l16  = lane & 15;
    const int koff = half * 2;
    const int aRow = m0 + l16;       // A row this lane feeds
    const int bCol = n0 + l16;       // B/C column this lane feeds

    const float* __restrict__ xrow = X   + (size_t)aRow * KDIM;
    const float* __restrict__ grow = AGG + (size_t)aRow * KDIM;

    v8f c;
    const float bv = bias[bCol];
#pragma unroll
    for (int r = 0; r < 8; ++r) c[r] = bv;

#pragma unroll 4
    for (int k = 0; k < KDIM; k += 4) {
        const int ka = k + koff;
        v2f a, b;
        a.x = xrow[ka]     + grow[ka];
        a.y = xrow[ka + 1] + grow[ka + 1];
        b.x = W[(size_t)(ka)     * NDIM + bCol];
        b.y = W[(size_t)(ka + 1) * NDIM + bCol];
        // (neg_a, A, neg_b, B, c_mod, C, reuse_a, reuse_b)
        c = __builtin_amdgcn_wmma_f32_16x16x4_f32(false, a, false, b,
                                                  (short)0, c, false, false);
    }

    if (RELU) {
#pragma unroll
        for (int r = 0; r < 8; ++r) c[r] = c[r] > 0.f ? c[r] : 0.f;
    }

#pragma unroll
    for (int r = 0; r < 8; ++r) {
        const int row = m0 + half * 8 + r;
        OUT[(size_t)row * NDIM + bCol] = c[r];
    }
}

// ---------------------------------------------------------------------------
// In-place log-softmax over 64 columns.  One wave per row, 2 values/lane,
// wave32 shuffle reductions for max and sum(exp).
// ---------------------------------------------------------------------------
__global__ __launch_bounds__(256) void logsoftmax_kernel(float* __restrict__ out) {
    int wave = blockIdx.x * (blockDim.x >> 5) + (threadIdx.x >> 5);
    if (wave >= N_NODES) return;
    int lane = threadIdx.x & 31;

    float* row = out + (size_t)wave * OUT_CH;
    float v0 = row[lane];
    float v1 = row[lane + 32];

    float m = fmaxf(v0, v1);
#pragma unroll
    for (int off = 16; off > 0; off >>= 1)
        m = fmaxf(m, __shfl_xor(m, off, 32));

    float e = __expf(v0 - m) + __expf(v1 - m);
#pragma unroll
    for (int off = 16; off > 0; off >>= 1)
        e += __shfl_xor(e, off, 32);

    float lse = m + __logf(e);
    row[lane]      = v0 - lse;
    row[lane + 32] = v1 - lse;
}

// ---------------------------------------------------------------------------
extern "C" void kernel_launch(void* const* d_in, const int* in_sizes, int n_in,
                              void* d_out, int out_size, void* d_ws, size_t ws_size,
                              hipStream_t stream) {
    const float* x    = (const float*)d_in[0];
    const int*   edge = (const int*)  d_in[1];   // [2, N_EDGES]
    const float* W1   = (const float*)d_in[2];
    const float* b1   = (const float*)d_in[3];
    const float* W2   = (const float*)d_in[4];
    const float* b2   = (const float*)d_in[5];
    const float* W3   = (const float*)d_in[6];
    const float* b3   = (const float*)d_in[7];
    float*       out  = (float*)d_out;           // [N_NODES, OUT_CH]

    const int* src = edge;
    const int* dst = edge + N_EDGES;

    // Workspace layout: agg | h1 | h2  (each N_NODES*HID f32 = 51.2 MB)
    const size_t featN = (size_t)N_NODES * HID;  // 12.8M floats
    float* agg = (float*)d_ws;
    float* h1  = agg + featN;
    float* h2  = h1  + featN;

    const dim3 blk(256);
    const dim3 zeroGrid((unsigned)(featN / 4 / 256));        // 12500
    const dim3 edgeGrid((N_EDGES + 7) / 8);                  // 200000 (8 waves/block)
    const dim3 gemmGrid128((N_NODES / 16) * (HID / 16) / 8); // 6250
    const dim3 gemmGrid64((N_NODES / 16) * (OUT_CH / 16) / 8); // 3125
    const dim3 smGrid((N_NODES + 7) / 8);                    // 12500

    // ---- Layer 1: h1 = relu((x + agg(x)) @ W1 + b1)
    zero_kernel<<<zeroGrid, blk, 0, stream>>>(agg, featN);
    scatter_kernel<<<edgeGrid, blk, 0, stream>>>(x, src, dst, agg);
    gin_gemm_kernel<IN_CH, HID, true><<<gemmGrid128, blk, 0, stream>>>(x, agg, W1, b1, h1);

    // ---- Layer 2: h2 = relu((h1 + agg(h1)) @ W2 + b2)
    zero_kernel<<<zeroGrid, blk, 0, stream>>>(agg, featN);
    scatter_kernel<<<edgeGrid, blk, 0, stream>>>(h1, src, dst, agg);
    gin_gemm_kernel<HID, HID, true><<<gemmGrid128, blk, 0, stream>>>(h1, agg, W2, b2, h2);

    // ---- Layer 3: out = (h2 + agg(h2)) @ W3 + b3, then log_softmax rows
    zero_kernel<<<zeroGrid, blk, 0, stream>>>(agg, featN);
    scatter_kernel<<<edgeGrid, blk, 0, stream>>>(h2, src, dst, agg);
    gin_gemm_kernel<HID, OUT_CH, false><<<gemmGrid64, blk, 0, stream>>>(h2, agg, W3, b3, out);
    logsoftmax_kernel<<<smGrid, blk, 0, stream>>>(out);
}